// ModelB_67233418051683
// MI455X (gfx1250) — compile-verified
//
#include <hip/hip_runtime.h>

#define S21 21
#define SS 441          // 21*21
#define OUTSTRIDE 1765  // 1 + 441*4

typedef __attribute__((ext_vector_type(2))) float v2f;
typedef __attribute__((ext_vector_type(8))) float v8f;

__device__ __forceinline__ int imin(int a, int b) { return a < b ? a : b; }
__device__ __forceinline__ int imax(int a, int b) { return a > b ? a : b; }

__global__ __launch_bounds__(256) void fcos_loss_kernel(
    const float* __restrict__ xff,   // [B,4,21,21]
    const float* __restrict__ cls3,  // [B,1,21,21]
    const float* __restrict__ bbox,  // [B,4]
    float* __restrict__ out,         // [B,1765]
    int B)
{
    const int lane = threadIdx.x & 31;
    const int wave = threadIdx.x >> 5;
    const int b = blockIdx.x * 8 + wave;
    if (b >= B) return;   // wave-uniform: EXEC stays all-ones for live waves

    // ---- per-element scalars (uniform across the wave) ----
    const float bx1 = bbox[(size_t)b * 4 + 0];
    const float by1 = bbox[(size_t)b * 4 + 1];
    const float bx2 = bbox[(size_t)b * 4 + 2];
    const float by2 = bbox[(size_t)b * 4 + 3];

    // idx2 = trunc((bbox-63)/8) ; idx = clip(idx2, 0, 20). /8 == *0.125 exactly.
    const int ix1 = (int)((bx1 - 63.0f) * 0.125f);
    const int iy1 = (int)((by1 - 63.0f) * 0.125f);
    const int ix2 = (int)((bx2 - 63.0f) * 0.125f);
    const int iy2 = (int)((by2 - 63.0f) * 0.125f);

    const int cx1 = imin(imax(ix1, 0), S21 - 1);
    const int cy1 = imin(imax(iy1, 0), S21 - 1);
    const int cx2 = imin(imax(ix2, 0), S21 - 1);
    const int cy2 = imin(imax(iy2, 0), S21 - 1);

    // dilated weightxff box (uses unclipped idx2; h2,w2 >= 0 for these inputs)
    const int w2 = ix2 - ix1;
    const int h2 = iy2 - iy1;
    const int lo_r = imax(0, iy1 - h2 / 2);
    const int hi_r = imin(S21, iy2 + 1 + h2 / 2);
    const int lo_c = imax(0, ix1 - w2 / 2);
    const int hi_c = imin(S21, ix2 + 1 + w2 / 2);

    // closed-form denominators (mask cell counts)
    const float cnt0 = (float)((cy2 - cy1 + 1) * (cx2 - cx1 + 1));
    const float cnt1 = (float)(imax(0, hi_r - lo_r) * imax(0, hi_c - lo_c));

    const float* xb = xff + (size_t)b * (4 * SS);
    const float* cb = cls3 + (size_t)b * SS;
    float* ob = out + (size_t)b * OUTSTRIDE;

    float s0 = 0.0f;  // sum weightcls3 * |cls3 - weightcls33|
    float s1 = 0.0f;  // sum weightxff  * sum_c |xff_c - label_c|

    // ---- streaming loop: lane-contiguous global accesses (coalesced) ----
    for (int i = lane; i < SS; i += 32) {
        const int y = i / S21;
        const int x = i - y * S21;
        const float gx = 8.0f * (float)x + 63.0f;
        const float gy = 8.0f * (float)y + 63.0f;

        const float x0 = __builtin_nontemporal_load(xb + i);
        const float x1 = __builtin_nontemporal_load(xb + SS + i);
        const float x2 = __builtin_nontemporal_load(xb + 2 * SS + i);
        const float x3 = __builtin_nontemporal_load(xb + 3 * SS + i);
        const float cv = __builtin_nontemporal_load(cb + i);

        // anchors pr: [gx-143*x0, gy-143*x2, gx+143*x1, gy+143*x3]
        float* po = ob + 1 + 4 * i;
        __builtin_nontemporal_store(gx - 143.0f * x0, po + 0);
        __builtin_nontemporal_store(gy - 143.0f * x2, po + 1);
        __builtin_nontemporal_store(gx + 143.0f * x1, po + 2);
        __builtin_nontemporal_store(gy + 143.0f * x3, po + 3);

        // cls centerness-style loss term
        if ((y >= cy1) & (y <= cy2) & (x >= cx1) & (x <= cx2)) {
            const float a = (float)(y - cy1), bb = (float)(cy2 - y);
            const float l1 = fminf(a, bb) / (fmaxf(a, bb) + 1e-4f);
            const float c = (float)(x - cx1), dd = (float)(cx2 - x);
            const float l2 = fminf(c, dd) / (fmaxf(c, dd) + 1e-4f);
            const float w33 = sqrtf(fmaxf(l1 * l2, 0.0f));
            s0 += fabsf(cv - w33);
        }

        // shape (ltrb regression) loss term
        if ((y >= lo_r) & (y < hi_r) & (x >= lo_c) & (x < hi_c)) {
            const float lb0 = (gx - bx1) / 143.0f;
            const float lb1 = (bx2 - gx) / 143.0f;
            const float lb2 = (gy - by1) / 143.0f;
            const float lb3 = (by2 - gy) / 143.0f;
            s1 += fabsf(x0 - lb0) + fabsf(x1 - lb1) +
                  fabsf(x2 - lb2) + fabsf(x3 - lb3);
        }
    }

    // ---- cross-lane reduction via V_WMMA_F32_16X16X4_F32 ----
    // A = (s, 0) per lane, B = all-ones. D[m][n] = rowsum(m); sum of all row
    // sums == sum of all A storage slots == sum over lanes of s, independent
    // of the exact A/B (lane,K) mapping. C/D layout: VGPR j holds rows j, j+8
    // across lane halves, so per-lane sum of d[0..7] + xor-16 fold = total.
    v8f cz = {0.f, 0.f, 0.f, 0.f, 0.f, 0.f, 0.f, 0.f};
    v2f ones; ones[0] = 1.0f; ones[1] = 1.0f;
    v2f av;   av[0] = s0;     av[1] = 0.0f;
    v8f d0 = __builtin_amdgcn_wmma_f32_16x16x4_f32(
        false, av, false, ones, (short)0, cz, false, false);
    av[0] = s1;
    v8f d1 = __builtin_amdgcn_wmma_f32_16x16x4_f32(
        false, av, false, ones, (short)0, cz, false, false);

    float t0 = d0[0] + d0[1] + d0[2] + d0[3] + d0[4] + d0[5] + d0[6] + d0[7];
    float t1 = d1[0] + d1[1] + d1[2] + d1[3] + d1[4] + d1[5] + d1[6] + d1[7];
    t0 += __shfl_xor(t0, 16, 32);
    t1 += __shfl_xor(t1, 16, 32);

    if (lane == 0) {
        ob[0] = t0 / (cnt0 + 1e-6f) + t1 / (cnt1 + 1e-6f);
    }
}

extern "C" void kernel_launch(void* const* d_in, const int* in_sizes, int n_in,
                              void* d_out, int out_size, void* d_ws, size_t ws_size,
                              hipStream_t stream) {
    const float* xff  = (const float*)d_in[0];
    const float* cls3 = (const float*)d_in[1];
    const float* bbox = (const float*)d_in[2];
    float* out = (float*)d_out;

    const int B = in_sizes[2] / 4;          // bbox is [B,4]
    const int blocks = (B + 7) / 8;         // 8 waves (8 batch elems) / 256-thread block
    fcos_loss_kernel<<<blocks, 256, 0, stream>>>(xff, cls3, bbox, out, B);
}